// Router_35167192220523
// MI455X (gfx1250) — compile-verified
//
#include <hip/hip_runtime.h>
#include <math.h>

typedef __attribute__((ext_vector_type(2))) float v2f;
typedef __attribute__((ext_vector_type(4))) float v4f;
typedef __attribute__((ext_vector_type(8))) float v8f;

#define HIDDEN 1024
#define NUM_EXPERTS 16
#define TOKENS_PER_WAVE 16
#define WAVES_PER_BLOCK 8
#define THREADS_PER_BLOCK (WAVES_PER_BLOCK * 32)

__global__ __launch_bounds__(THREADS_PER_BLOCK)
void router_wmma_kernel(const float* __restrict__ X,
                        const float* __restrict__ W,
                        const float* __restrict__ bias,
                        float* __restrict__ out,
                        int n_tokens) {
    // per-wave logits staging: [wave][token-in-tile][expert]
    __shared__ float lds_logits[WAVES_PER_BLOCK][TOKENS_PER_WAVE][NUM_EXPERTS];

    const int lane   = threadIdx.x & 31;
    const int wave   = threadIdx.x >> 5;
    const int laneLo = lane & 15;   // token row (A) / expert col (B,C)
    const int laneHi = lane >> 4;   // K-half selector for A/B fragments

    const int tile      = blockIdx.x * WAVES_PER_BLOCK + wave;
    const int tokenBase = tile * TOKENS_PER_WAVE;
    if (tokenBase >= n_tokens) return;   // wave-uniform: whole wave exits

    // A fragment stream: lane holds X[tokenBase+laneLo][4j + 2*laneHi .. +1]
    const float* aPtr = X + (size_t)(tokenBase + laneLo) * HIDDEN + 2 * laneHi;
    // B fragment stream: B = W^T, so lane holds W[laneLo][4j + 2*laneHi .. +1]
    const float* bPtr = W + (size_t)laneLo * HIDDEN + 2 * laneHi;

    v8f acc = {};
#pragma unroll 8
    for (int j = 0; j < HIDDEN / 4; ++j) {
        v2f a = *(const v2f*)(aPtr + 4 * j);
        v2f b = *(const v2f*)(bPtr + 4 * j);
        // D = A(16x4) * B(4x16) + C ; fp32 WMMA
        acc = __builtin_amdgcn_wmma_f32_16x16x4_f32(
            /*neg_a=*/false, a, /*neg_b=*/false, b,
            /*c_mod=*/(short)0, acc, /*reuse_a=*/false, /*reuse_b=*/false);
    }

    // C/D layout: vgpr i, lane -> D[M = i + 8*laneHi][N = laneLo]
    const float bv = bias[laneLo];                       // expert bias (zeros here, but honor reference)
    float* logitsOut = out + (size_t)n_tokens * NUM_EXPERTS;  // second output tensor
#pragma unroll
    for (int i = 0; i < 8; ++i) {
        const int m = i + 8 * laneHi;
        const float v = acc[i] + bv;
        lds_logits[wave][m][laneLo] = v;
        logitsOut[(size_t)(tokenBase + m) * NUM_EXPERTS + laneLo] = v;
    }
    __syncthreads();

    // One lane per token: top-2 -> renormalized softmax weights -> dense sparse row.
    // softmax-then-top2-renorm == 2-way softmax over the top-2 logits.
    if (laneHi == 0) {
        const float* lv = lds_logits[wave][laneLo];
        int i1 = 0; float l1 = lv[0];
#pragma unroll
        for (int e = 1; e < NUM_EXPERTS; ++e) {
            float x = lv[e];
            if (x > l1) { l1 = x; i1 = e; }
        }
        int i2 = -1; float l2 = -INFINITY;
#pragma unroll
        for (int e = 0; e < NUM_EXPERTS; ++e) {
            float x = lv[e];
            if (e != i1 && x > l2) { l2 = x; i2 = e; }
        }
        const float w1 = 1.0f / (1.0f + __expf(l2 - l1));
        const float w2 = 1.0f - w1;

        float* srow = out + (size_t)(tokenBase + laneLo) * NUM_EXPERTS;
#pragma unroll
        for (int k = 0; k < 4; ++k) {
            v4f v = {};
#pragma unroll
            for (int c = 0; c < 4; ++c) {
                const int e = 4 * k + c;
                v[c] = (e == i1) ? w1 : ((e == i2) ? w2 : 0.0f);
            }
            *(v4f*)(srow + 4 * k) = v;
        }
    }
}

extern "C" void kernel_launch(void* const* d_in, const int* in_sizes, int n_in,
                              void* d_out, int out_size, void* d_ws, size_t ws_size,
                              hipStream_t stream) {
    const float* X    = (const float*)d_in[0];   // [B*S, 1024] fp32
    const float* W    = (const float*)d_in[1];   // [16, 1024]  fp32
    const float* bias = (const float*)d_in[2];   // [16]        fp32
    float* out = (float*)d_out;                  // [sparse | logits], each n_tokens*16

    const int n_tokens = in_sizes[0] / HIDDEN;   // 32768
    const int tiles  = (n_tokens + TOKENS_PER_WAVE - 1) / TOKENS_PER_WAVE;
    const int blocks = (tiles + WAVES_PER_BLOCK - 1) / WAVES_PER_BLOCK;

    router_wmma_kernel<<<blocks, THREADS_PER_BLOCK, 0, stream>>>(X, W, bias, out, n_tokens);
}